// QConv1D_79216376807791
// MI455X (gfx1250) — compile-verified
//
#include <hip/hip_runtime.h>
#include <hip/hip_bf16.h>

typedef __attribute__((ext_vector_type(2))) float v2f;
typedef __attribute__((ext_vector_type(8))) float v8f;

#define NQ      6
#define DIM     64
#define OUT_CH  16
#define VLAYERS 2
#define Bn      16
#define Lx      1024
#define L_OUT   1022
#define NP      16352      /* Bn * L_OUT */
#define KTOT    128        /* 2*DIM real rows */
#define MTOT    2048       /* OUT_CH * 128 */

// ---------------------------------------------------------------------------
// Kernel 1: build the stacked real matrix A (2048 x 128) from weight.
// Thread (oc, col) evolves basis state e_col through 2 layers of 6 RY + CZ.
// A[oc*128 + d][k]      = U_r[d][k]   (k<64)  |  -U_i[d][k-64] (k>=64)
// A[oc*128 + 64+d][k]   = U_i[d][k]   (k<64)  |   U_r[d][k-64] (k>=64)
// ---------------------------------------------------------------------------
__global__ void build_u_kernel(const float* __restrict__ w, float* __restrict__ U) {
    int gid = blockIdx.x * blockDim.x + threadIdx.x;
    if (gid >= OUT_CH * DIM) return;
    int oc  = gid >> 6;
    int col = gid & 63;

    float sr[DIM], si[DIM];
#pragma unroll
    for (int d = 0; d < DIM; ++d) { sr[d] = (d == col) ? 1.0f : 0.0f; si[d] = 0.0f; }

#pragma unroll
    for (int l = 0; l < VLAYERS; ++l) {
#pragma unroll
        for (int q = 0; q < NQ; ++q) {
            float th = w[oc * (NQ * VLAYERS) + q * VLAYERS + l];
            float ch = cosf(0.5f * th);
            float sh = sinf(0.5f * th);
            const int stride = 1 << (5 - q);
#pragma unroll
            for (int i = 0; i < DIM; ++i) {
                if ((i & stride) == 0) {
                    int j = i | stride;
                    float a0r = sr[i], a0i = si[i];
                    float a1r = sr[j], a1i = si[j];
                    sr[i] = ch * a0r - sh * a1r;
                    si[i] = ch * a0i - sh * a1i;
                    sr[j] = sh * a0r + ch * a1r;
                    si[j] = sh * a0i + ch * a1i;
                }
            }
        }
        // ring of CZ gates: sign = (-1)^(sum of adjacent-bit products on a 6-ring)
#pragma unroll
        for (int d = 0; d < DIM; ++d) {
            int rot = ((d << 1) | (d >> 5)) & 63;
            if (__popc(d & rot) & 1) { sr[d] = -sr[d]; si[d] = -si[d]; }
        }
    }

#pragma unroll
    for (int d = 0; d < DIM; ++d) {
        int r1 = oc * KTOT + d;
        int r2 = r1 + DIM;
        U[r1 * KTOT + col]       =  sr[d];
        U[r1 * KTOT + DIM + col] = -si[d];
        U[r2 * KTOT + col]       =  si[d];
        U[r2 * KTOT + DIM + col] =  sr[d];
    }
}

// ---------------------------------------------------------------------------
// Kernel 2: build S (128 x NP): rows 0..63 = (1/8)cos(phi_d), 64..127 = sin.
// phi_d(n) = sum_q x_patch[n][q] * (bit_q(d) - 0.5), bit_q(d) = (d>>(5-q))&1,
// patch q = c*3 + k -> x[b][c][lo+k].
// ---------------------------------------------------------------------------
__global__ void build_s_kernel(const float* __restrict__ x, float* __restrict__ S) {
    int n = blockIdx.x * blockDim.x + threadIdx.x;
    int d = blockIdx.y;
    if (n >= NP) return;
    int b  = n / L_OUT;
    int lo = n - b * L_OUT;
    const float* xb = x + b * 2 * Lx;

    float phi = 0.0f;
#pragma unroll
    for (int c = 0; c < 2; ++c) {
#pragma unroll
        for (int k = 0; k < 3; ++k) {
            int q = c * 3 + k;
            float bit = (float)((d >> (5 - q)) & 1) - 0.5f;
            phi += xb[c * Lx + lo + k] * bit;
        }
    }
    float sn, cs;
    sincosf(phi, &sn, &cs);
    S[d * NP + n]         = 0.125f * cs;
    S[(DIM + d) * NP + n] = 0.125f * sn;
}

// ---------------------------------------------------------------------------
// Kernel 3: Y = A x S via V_WMMA_F32_16X16X4_F32, fused Z-weighted |.|^2
// reduction.  Block = (oc, 16-column strip); 8 waves cover the 128 rows.
// A 16x4 frag : lane<16 -> {A[M=lid][k0], A[lid][k0+1]}, lane>=16 -> K=k0+2,+3
// B 4x16 frag : lane<16 -> {B[k0][lid],   B[k0+1][lid]}, lane>=16 -> K=k0+2,+3
// C/D         : lane<16 -> rows m+0..7,   lane>=16 -> rows m+8..15
// ---------------------------------------------------------------------------
__global__ __launch_bounds__(256)
void qgemm_kernel(const float* __restrict__ A, const float* __restrict__ S,
                  float* __restrict__ out) {
    const int nb   = blockIdx.x << 4;   // column base (0..16336)
    const int oc   = blockIdx.y;
    const int tid  = threadIdx.x;
    const int w    = tid >> 5;          // wave 0..7 -> row tile
    const int lane = tid & 31;
    const int half = lane >> 4;
    const int lid  = lane & 15;

    __shared__ float Bs[KTOT][16];      // 8 KB B strip, shared by all 8 waves
    __shared__ float red[8][16];

    for (int i = tid; i < KTOT * 16; i += 256) {
        int k = i >> 4, c = i & 15;
        Bs[k][c] = S[k * NP + nb + c];
    }
    __syncthreads();

    const float* Arow = A + (size_t)(oc * KTOT + (w << 4) + lid) * KTOT;

    v8f acc = {};
#pragma unroll
    for (int k0 = 0; k0 < KTOT; k0 += 4) {
        const int kk = k0 + (half << 1);
        v2f a, b;
        a.x = Arow[kk];
        a.y = Arow[kk + 1];
        b.x = Bs[kk][lid];
        b.y = Bs[kk + 1][lid];
        acc = __builtin_amdgcn_wmma_f32_16x16x4_f32(
            false, a, false, b, (short)0, acc, false, false);
    }

    // exp contribution: sum_r Z(r&63) * Y[r]^2 for this wave's 16 rows
    float p = 0.0f;
#pragma unroll
    for (int v = 0; v < 8; ++v) {
        int r = (w << 4) + (half << 3) + v;     // row within 0..127
        int d = r & 63;
        float z = (__popc(d) & 1) ? -1.0f : 1.0f;
        p = fmaf(z * acc[v], acc[v], p);
    }
    p += __shfl_xor(p, 16, 32);                 // combine lane halves (wave32)
    if (lane < 16) red[w][lane] = p;
    __syncthreads();

    if (tid < 16) {
        float s = 0.0f;
#pragma unroll
        for (int wv = 0; wv < 8; ++wv) s += red[wv][tid];
        // reference reshape scramble: out_flat = oc*NP + (n%16)*L_OUT + n/16
        out[oc * NP + tid * L_OUT + blockIdx.x] = s;
    }
}

// ---------------------------------------------------------------------------
extern "C" void kernel_launch(void* const* d_in, const int* in_sizes, int n_in,
                              void* d_out, int out_size, void* d_ws, size_t ws_size,
                              hipStream_t stream) {
    const float* x      = (const float*)d_in[0];   // (16, 2, 1024) f32
    const float* weight = (const float*)d_in[1];   // (16, 6, 2)   f32
    float* out = (float*)d_out;                    // 261632 f32

    float* U = (float*)d_ws;                       // 2048*128 f32 = 1 MB
    float* S = U + (size_t)MTOT * KTOT;            // 128*16352 f32 = 8.37 MB

    build_u_kernel<<<dim3(4), dim3(256), 0, stream>>>(weight, U);
    build_s_kernel<<<dim3((NP + 255) / 256, DIM), dim3(256), 0, stream>>>(x, S);
    qgemm_kernel<<<dim3(NP / 16, OUT_CH), dim3(256), 0, stream>>>(U, S, out);
}